// ThetaLayer_77953656422942
// MI455X (gfx1250) — compile-verified
//
#include <hip/hip_runtime.h>
#include <cstdint>
#include <cstddef>

// ---------------------------------------------------------------------------
// Problem constants (match reference)
// ---------------------------------------------------------------------------
#define Bb 64
#define Kk 16
#define Nn 64
#define Ll 512
#define Mm 512
#define Jj 5

typedef __attribute__((ext_vector_type(2))) float v2f;
typedef __attribute__((ext_vector_type(8))) float v8f;

// V_WMMA_F32_16X16X4_F32: D(16x16 f32) = A(16x4 f32) * B(4x16 f32) + C
// 8-arg form: (neg_a, A, neg_b, B, c_mod, C, reuse_a, reuse_b)
__device__ __forceinline__ v8f wmma4(v2f a, v2f b, v8f c) {
  return __builtin_amdgcn_wmma_f32_16x16x4_f32(false, a, false, b, (short)0, c,
                                               false, false);
}
__device__ __forceinline__ v8f z8() {
  v8f z = {0.f, 0.f, 0.f, 0.f, 0.f, 0.f, 0.f, 0.f};
  return z;
}

// Workspace layout (units: float2 complex elements)
static const size_t OFF_Y   = 0;                                   // B*N*N
static const size_t OFF_A2  = OFF_Y   + (size_t)Bb * Nn * Nn;      // B*L*N
static const size_t OFF_T   = OFF_A2  + (size_t)Bb * Ll * Nn;      // B*K*M
static const size_t OFF_XTH = OFF_T   + (size_t)Bb * Kk * Mm;      // B*N*K
static const size_t OFF_D   = OFF_XTH + (size_t)Bb * Nn * Kk;      // B*L
static const size_t OFF_G   = OFF_D   + (size_t)Bb * Ll;           // B*L*L

// ---------------------------------------------------------------------------
// Kernel 1: Y = X @ X^H   (per batch: N x N, K-loop over M)
//   Y_re = Xre Xre^T + Xim Xim^T ;  Y_im = Xim Xre^T - Xre Xim^T
// ---------------------------------------------------------------------------
__global__ void __launch_bounds__(256)
k_xxh(const float* __restrict__ Xre, const float* __restrict__ Xim,
      float2* __restrict__ Yws) {
  const int lane = threadIdx.x & 31;
  const int tile = blockIdx.x * 8 + (threadIdx.x >> 5);  // 0..1023
  const int b  = tile >> 4;
  const int t  = tile & 15;
  const int r0 = (t >> 2) << 4;
  const int c0 = (t & 3) << 4;
  const int ml = lane & 15;
  const int kh = lane >> 4;

  const float* ar = Xre + (size_t)b * Nn * Mm + (size_t)(r0 + ml) * Mm;
  const float* ai = Xim + (size_t)b * Nn * Mm + (size_t)(r0 + ml) * Mm;
  const float* br = Xre + (size_t)b * Nn * Mm + (size_t)(c0 + ml) * Mm;
  const float* bi = Xim + (size_t)b * Nn * Mm + (size_t)(c0 + ml) * Mm;

  v8f p1 = z8(), p2 = z8(), p3 = z8(), p4 = z8();
  for (int k0 = 0; k0 < Mm; k0 += 4) {
    const int ka = k0 + 2 * kh;
    v2f are = *(const v2f*)(ar + ka);
    v2f aim = *(const v2f*)(ai + ka);
    v2f bre = *(const v2f*)(br + ka);
    v2f bim = *(const v2f*)(bi + ka);
    p1 = wmma4(are, bre, p1);
    p2 = wmma4(aim, bim, p2);
    p3 = wmma4(aim, bre, p3);
    p4 = wmma4(are, bim, p4);
  }
  v8f yre = p1 + p2;
  v8f yim = p3 - p4;
  float2* yb = Yws + (size_t)b * Nn * Nn;
#pragma unroll
  for (int r = 0; r < 8; ++r) {
    const int row = r0 + r + 8 * kh;
    yb[(size_t)row * Nn + (c0 + ml)] = make_float2(yre[r], yim[r]);
  }
}

// ---------------------------------------------------------------------------
// Kernel 2: A2 = H_br @ Y   (per batch: L x N, K-loop over N)
// ---------------------------------------------------------------------------
__global__ void __launch_bounds__(256)
k_a2(const float* __restrict__ Hbr_re, const float* __restrict__ Hbr_im,
     const float2* __restrict__ Yws, float2* __restrict__ A2ws) {
  const int lane = threadIdx.x & 31;
  const int tile = blockIdx.x * 8 + (threadIdx.x >> 5);  // 0..8191
  const int b  = tile >> 7;
  const int t  = tile & 127;                 // 32 x 4 tiles
  const int r0 = (t >> 2) << 4;
  const int c0 = (t & 3) << 4;
  const int ml = lane & 15;
  const int kh = lane >> 4;

  const float*  hr = Hbr_re + (size_t)b * Ll * Nn + (size_t)(r0 + ml) * Nn;
  const float*  hi = Hbr_im + (size_t)b * Ll * Nn + (size_t)(r0 + ml) * Nn;
  const float2* yb = Yws + (size_t)b * Nn * Nn;

  v8f p1 = z8(), p2 = z8(), p3 = z8(), p4 = z8();
  for (int k0 = 0; k0 < Nn; k0 += 4) {
    const int ka = k0 + 2 * kh;
    v2f are = *(const v2f*)(hr + ka);
    v2f aim = *(const v2f*)(hi + ka);
    float2 y0 = yb[(size_t)ka * Nn + c0 + ml];
    float2 y1 = yb[(size_t)(ka + 1) * Nn + c0 + ml];
    v2f bre = {y0.x, y1.x};
    v2f bim = {y0.y, y1.y};
    p1 = wmma4(are, bre, p1);
    p2 = wmma4(aim, bim, p2);
    p3 = wmma4(are, bim, p3);
    p4 = wmma4(aim, bre, p4);
  }
  v8f cre = p1 - p2;   // re = are*bre - aim*bim
  v8f cim = p3 + p4;   // im = are*bim + aim*bre
  float2* a2 = A2ws + (size_t)b * Ll * Nn;
#pragma unroll
  for (int r = 0; r < 8; ++r) {
    const int row = r0 + r + 8 * kh;
    a2[(size_t)row * Nn + (c0 + ml)] = make_float2(cre[r], cim[r]);
  }
}

// ---------------------------------------------------------------------------
// Kernel 3 (fused): G[l,m] = Bm[l,m] * Cm[m,l]
//   CmT[l,m] = sum_n conj(Hbr[l,n]) * A2[m,n]          (K-loop N=64)
//   Bm[l,m]  = sum_k conj(Hru[k,l]) * Hru[k,m]         (K-loop K=16)
// ---------------------------------------------------------------------------
__global__ void __launch_bounds__(256)
k_gfuse(const float* __restrict__ Hbr_re, const float* __restrict__ Hbr_im,
        const float* __restrict__ Hru_re, const float* __restrict__ Hru_im,
        const float2* __restrict__ A2ws, float2* __restrict__ Gws) {
  const int lane = threadIdx.x & 31;
  const int tile = blockIdx.x * 8 + (threadIdx.x >> 5);  // 0..65535
  const int b  = tile >> 10;
  const int t  = tile & 1023;                // 32 x 32 tiles
  const int r0 = (t >> 5) << 4;              // l-tile
  const int c0 = (t & 31) << 4;              // m-tile
  const int ml = lane & 15;
  const int kh = lane >> 4;

  // ---- CmT part: A = conj(Hbr[r0.., :]) , B[n,m] = A2[c0+m, n]
  const float*  hr  = Hbr_re + (size_t)b * Ll * Nn + (size_t)(r0 + ml) * Nn;
  const float*  hi  = Hbr_im + (size_t)b * Ll * Nn + (size_t)(r0 + ml) * Nn;
  const float2* a2r = A2ws + (size_t)b * Ll * Nn + (size_t)(c0 + ml) * Nn;

  v8f q1 = z8(), q2 = z8(), q3 = z8(), q4 = z8();
  for (int k0 = 0; k0 < Nn; k0 += 4) {
    const int ka = k0 + 2 * kh;
    v2f are = *(const v2f*)(hr + ka);
    v2f aim = *(const v2f*)(hi + ka);
    float2 c0v = a2r[ka];
    float2 c1v = a2r[ka + 1];
    v2f bre = {c0v.x, c1v.x};
    v2f bim = {c0v.y, c1v.y};
    q1 = wmma4(are, bre, q1);
    q2 = wmma4(aim, bim, q2);
    q3 = wmma4(are, bim, q3);
    q4 = wmma4(aim, bre, q4);
  }
  v8f cmt_re = q1 + q2;   // conj(A): re = ar*br + ai*bi
  v8f cmt_im = q3 - q4;   //          im = ar*bi - ai*br

  // ---- Bm part: A[l,k] = conj(Hru[k, r0+l]) , B[k,m] = Hru[k, c0+m]
  const float* ur = Hru_re + (size_t)b * Kk * Ll;
  const float* ui = Hru_im + (size_t)b * Kk * Ll;
  v8f s1 = z8(), s2 = z8(), s3 = z8(), s4 = z8();
  for (int k0 = 0; k0 < Kk; k0 += 4) {
    const int ka = k0 + 2 * kh;
    v2f are = {ur[(size_t)ka * Ll + r0 + ml], ur[(size_t)(ka + 1) * Ll + r0 + ml]};
    v2f aim = {ui[(size_t)ka * Ll + r0 + ml], ui[(size_t)(ka + 1) * Ll + r0 + ml]};
    v2f bre = {ur[(size_t)ka * Ll + c0 + ml], ur[(size_t)(ka + 1) * Ll + c0 + ml]};
    v2f bim = {ui[(size_t)ka * Ll + c0 + ml], ui[(size_t)(ka + 1) * Ll + c0 + ml]};
    s1 = wmma4(are, bre, s1);
    s2 = wmma4(aim, bim, s2);
    s3 = wmma4(are, bim, s3);
    s4 = wmma4(aim, bre, s4);
  }
  v8f bm_re = s1 + s2;
  v8f bm_im = s3 - s4;

  v8f g_re = bm_re * cmt_re - bm_im * cmt_im;
  v8f g_im = bm_re * cmt_im + bm_im * cmt_re;

  float2* gb = Gws + (size_t)b * Ll * Ll;
#pragma unroll
  for (int r = 0; r < 8; ++r) {
    const int row = r0 + r + 8 * kh;
    gb[(size_t)row * Ll + (c0 + ml)] = make_float2(g_re[r], g_im[r]);
  }
}

// ---------------------------------------------------------------------------
// Kernel 4: T = H_bu @ X - S   (per batch: K x M, K-loop over N)
// ---------------------------------------------------------------------------
__global__ void __launch_bounds__(256)
k_tmat(const float* __restrict__ Hbu_re, const float* __restrict__ Hbu_im,
       const float* __restrict__ Xre, const float* __restrict__ Xim,
       const float* __restrict__ Sre, const float* __restrict__ Sim,
       float2* __restrict__ Tws) {
  const int lane = threadIdx.x & 31;
  const int tile = blockIdx.x * 8 + (threadIdx.x >> 5);  // 0..2047
  const int b  = tile >> 5;
  const int c0 = (tile & 31) << 4;
  const int ml = lane & 15;
  const int kh = lane >> 4;

  const float* ar = Hbu_re + (size_t)b * Kk * Nn + (size_t)ml * Nn;
  const float* ai = Hbu_im + (size_t)b * Kk * Nn + (size_t)ml * Nn;
  const float* xr = Xre + (size_t)b * Nn * Mm;
  const float* xi = Xim + (size_t)b * Nn * Mm;

  v8f p1 = z8(), p2 = z8(), p3 = z8(), p4 = z8();
  for (int k0 = 0; k0 < Nn; k0 += 4) {
    const int ka = k0 + 2 * kh;
    v2f are = *(const v2f*)(ar + ka);
    v2f aim = *(const v2f*)(ai + ka);
    v2f bre = {xr[(size_t)ka * Mm + c0 + ml], xr[(size_t)(ka + 1) * Mm + c0 + ml]};
    v2f bim = {xi[(size_t)ka * Mm + c0 + ml], xi[(size_t)(ka + 1) * Mm + c0 + ml]};
    p1 = wmma4(are, bre, p1);
    p2 = wmma4(aim, bim, p2);
    p3 = wmma4(are, bim, p3);
    p4 = wmma4(aim, bre, p4);
  }
  v8f tre = p1 - p2;
  v8f tim = p3 + p4;
  float2* tb = Tws + (size_t)b * Kk * Mm;
  const float* sr = Sre + (size_t)b * Kk * Mm;
  const float* si = Sim + (size_t)b * Kk * Mm;
#pragma unroll
  for (int r = 0; r < 8; ++r) {
    const int row = r + 8 * kh;  // K rows 0..15
    const size_t idx = (size_t)row * Mm + (c0 + ml);
    tb[idx] = make_float2(tre[r] - sr[idx], tim[r] - si[idx]);
  }
}

// ---------------------------------------------------------------------------
// Kernel 5: XTH = X @ T^H   (per batch: N x K, K-loop over M)
// ---------------------------------------------------------------------------
__global__ void __launch_bounds__(256)
k_xth(const float* __restrict__ Xre, const float* __restrict__ Xim,
      const float2* __restrict__ Tws, float2* __restrict__ XThws) {
  const int lane = threadIdx.x & 31;
  const int tile = blockIdx.x * 8 + (threadIdx.x >> 5);  // 0..255
  const int b  = tile >> 2;
  const int r0 = (tile & 3) << 4;
  const int ml = lane & 15;
  const int kh = lane >> 4;

  const float*  ar = Xre + (size_t)b * Nn * Mm + (size_t)(r0 + ml) * Mm;
  const float*  ai = Xim + (size_t)b * Nn * Mm + (size_t)(r0 + ml) * Mm;
  const float2* tb = Tws + (size_t)b * Kk * Mm + (size_t)ml * Mm;  // row = k-col

  v8f p1 = z8(), p2 = z8(), p3 = z8(), p4 = z8();
  for (int k0 = 0; k0 < Mm; k0 += 4) {
    const int ka = k0 + 2 * kh;
    v2f are = *(const v2f*)(ar + ka);
    v2f aim = *(const v2f*)(ai + ka);
    float2 t0 = tb[ka];
    float2 t1 = tb[ka + 1];
    v2f bre = {t0.x, t1.x};
    v2f bim = {t0.y, t1.y};   // conj(T) handled in combine
    p1 = wmma4(are, bre, p1);
    p2 = wmma4(aim, bim, p2);
    p3 = wmma4(aim, bre, p3);
    p4 = wmma4(are, bim, p4);
  }
  v8f ore = p1 + p2;   // conj(B): re = ar*br + ai*bi
  v8f oim = p3 - p4;   //          im = ai*br - ar*bi
  float2* ob = XThws + (size_t)b * Nn * Kk;
#pragma unroll
  for (int r = 0; r < 8; ++r) {
    const int row = r0 + r + 8 * kh;
    ob[(size_t)row * Kk + ml] = make_float2(ore[r], oim[r]);
  }
}

// ---------------------------------------------------------------------------
// Kernel 6: dconj[b,l] = conj( sum_n Hbr[l,n] * (sum_k XTH[n,k]*Hru[k,l]) )
// ---------------------------------------------------------------------------
__global__ void __launch_bounds__(512)
k_dvec(const float* __restrict__ Hbr_re, const float* __restrict__ Hbr_im,
       const float* __restrict__ Hru_re, const float* __restrict__ Hru_im,
       const float2* __restrict__ XThws, float2* __restrict__ Dws) {
  __shared__ float2 sxth[Nn * Kk];
  const int b = blockIdx.x;
  const int l = threadIdx.x;
  const float2* xb = XThws + (size_t)b * Nn * Kk;
  for (int i = threadIdx.x; i < Nn * Kk; i += 512) sxth[i] = xb[i];
  __syncthreads();

  float hre[Kk], him[Kk];
#pragma unroll
  for (int k = 0; k < Kk; ++k) {
    hre[k] = Hru_re[(size_t)b * Kk * Ll + (size_t)k * Ll + l];
    him[k] = Hru_im[(size_t)b * Kk * Ll + (size_t)k * Ll + l];
  }
  const float* br = Hbr_re + (size_t)b * Ll * Nn + (size_t)l * Nn;
  const float* bi = Hbr_im + (size_t)b * Ll * Nn + (size_t)l * Nn;
  float dre = 0.f, dim = 0.f;
  for (int n = 0; n < Nn; ++n) {
    float wre = 0.f, wim = 0.f;
#pragma unroll
    for (int k = 0; k < Kk; ++k) {
      float2 x = sxth[n * Kk + k];
      wre += x.x * hre[k] - x.y * him[k];
      wim += x.x * him[k] + x.y * hre[k];
    }
    float hr = br[n], hi = bi[n];
    dre += hr * wre - hi * wim;
    dim += hr * wim + hi * wre;
  }
  Dws[(size_t)b * Ll + l] = make_float2(dre, -dim);  // store conj(d)
}

// ---------------------------------------------------------------------------
// Kernel 7: theta init + all J gradient steps + output (one block per batch)
// ---------------------------------------------------------------------------
__global__ void __launch_bounds__(512)
k_iter(const float2* __restrict__ Gws, const float2* __restrict__ Dws,
       const float* __restrict__ th0re, const float* __restrict__ th0im,
       float* __restrict__ out) {
  __shared__ float2 th[Ll];
  const int b = blockIdx.x;
  const int l = threadIdx.x;

  float tr = th0re[(size_t)b * Ll + l];
  float ti = th0im[(size_t)b * Ll + l];
  float inv = 1.f / fmaxf(sqrtf(tr * tr + ti * ti), 1e-12f);
  float myr = tr * inv, myi = ti * inv;
  th[l] = make_float2(myr, myi);
  __syncthreads();

  const float alpha = fminf(fmaxf(log1pf(expf(0.1f)), 1e-6f), 1.0f);
  const float2 dc = Dws[(size_t)b * Ll + l];
  const float4* g4 =
      (const float4*)(Gws + ((size_t)b * Ll + l) * Ll);  // row l of G

  for (int j = 0; j < Jj; ++j) {
    float nr = 0.f, ni = 0.f;
    for (int m = 0; m < Ll / 2; ++m) {
      float4 g = g4[m];
      float2 t0 = th[2 * m];
      float2 t1 = th[2 * m + 1];
      nr += g.x * t0.x - g.y * t0.y;
      ni += g.x * t0.y + g.y * t0.x;
      nr += g.z * t1.x - g.w * t1.y;
      ni += g.z * t1.y + g.w * t1.x;
    }
    nr = 2.f * (nr + dc.x);
    ni = 2.f * (ni + dc.y);
    // proj = Re(nabla*conj(theta)) * theta ; t = theta - alpha*(nabla - proj)
    float s = nr * myr + ni * myi;
    float txr = myr - alpha * (nr - s * myr);
    float txi = myi - alpha * (ni - s * myi);
    float iv = 1.f / fmaxf(sqrtf(txr * txr + txi * txi), 1e-12f);
    myr = txr * iv;
    myi = txi * iv;
    __syncthreads();
    th[l] = make_float2(myr, myi);
    __syncthreads();
  }
  out[((size_t)b * Ll + l) * 2 + 0] = myr;
  out[((size_t)b * Ll + l) * 2 + 1] = myi;
}

// ---------------------------------------------------------------------------
extern "C" void kernel_launch(void* const* d_in, const int* in_sizes, int n_in,
                              void* d_out, int out_size, void* d_ws,
                              size_t ws_size, hipStream_t stream) {
  (void)in_sizes; (void)n_in; (void)out_size; (void)ws_size;
  const float* Hbu_re = (const float*)d_in[0];
  const float* Hbu_im = (const float*)d_in[1];
  const float* Hru_re = (const float*)d_in[2];
  const float* Hru_im = (const float*)d_in[3];
  const float* Hbr_re = (const float*)d_in[4];
  const float* Hbr_im = (const float*)d_in[5];
  const float* X_re   = (const float*)d_in[6];
  const float* X_im   = (const float*)d_in[7];
  const float* S_re   = (const float*)d_in[8];
  const float* S_im   = (const float*)d_in[9];
  const float* th0_re = (const float*)d_in[10];
  const float* th0_im = (const float*)d_in[11];
  float* out = (float*)d_out;

  float2* ws   = (float2*)d_ws;
  float2* Y    = ws + OFF_Y;
  float2* A2   = ws + OFF_A2;
  float2* T    = ws + OFF_T;
  float2* XTH  = ws + OFF_XTH;
  float2* D    = ws + OFF_D;
  float2* G    = ws + OFF_G;

  // G-branch
  k_xxh  <<<128,  256, 0, stream>>>(X_re, X_im, Y);
  k_a2   <<<1024, 256, 0, stream>>>(Hbr_re, Hbr_im, Y, A2);
  k_gfuse<<<8192, 256, 0, stream>>>(Hbr_re, Hbr_im, Hru_re, Hru_im, A2, G);
  // d-branch
  k_tmat <<<256,  256, 0, stream>>>(Hbu_re, Hbu_im, X_re, X_im, S_re, S_im, T);
  k_xth  <<<32,   256, 0, stream>>>(X_re, X_im, T, XTH);
  k_dvec <<<Bb,   512, 0, stream>>>(Hbr_re, Hbr_im, Hru_re, Hru_im, XTH, D);
  // iterations + output
  k_iter <<<Bb,   512, 0, stream>>>(G, D, th0_re, th0_im, out);
}